// SAGEDense_49357764166103
// MI455X (gfx1250) — compile-verified
//
#include <hip/hip_runtime.h>

typedef __attribute__((ext_vector_type(2))) float v2f;
typedef __attribute__((ext_vector_type(8))) float v8f;

#define D 128
#define LDS_STRIDE 132   // 128 + 4 pad: breaks 16-way bank conflict on column reads

// ---------------------------------------------------------------------------
// Zero-fill fp32 buffer
// ---------------------------------------------------------------------------
__global__ __launch_bounds__(256) void sage_zero_f32(float* __restrict__ p, long long n) {
    long long i = (long long)blockIdx.x * blockDim.x + threadIdx.x;
    if (i < n) p[i] = 0.0f;
}

// ---------------------------------------------------------------------------
// Y = relu(X @ W + b)   X:[N,128] W:[128,128] b:[128]
// Block = 256 threads = 8 waves. Block owns 16 rows; wave w owns cols [16w,16w+16).
// A panel (16x128) staged in LDS once, shared by all 8 waves.
// WMMA: V_WMMA_F32_16X16X4_F32, K stepped by 4, 32 wmma per tile.
// ---------------------------------------------------------------------------
__global__ __launch_bounds__(256) void sage_dense_relu_wmma(
    const float* __restrict__ X, const float* __restrict__ W,
    const float* __restrict__ bias, float* __restrict__ Y, int N)
{
    __shared__ float As[16 * LDS_STRIDE];

    const int row0 = blockIdx.x << 4;              // 16 rows per block (N % 16 == 0)
    const int t    = threadIdx.x;

    // Stage A: 16x128 fp32 = 2048 floats, 256 threads x 2 float4 each, coalesced.
    const float4* Xv = (const float4*)(X + (size_t)row0 * D);
#pragma unroll
    for (int i = 0; i < 2; ++i) {
        int e4 = t * 2 + i;                        // float4 index 0..511
        int e  = e4 * 4;                           // element index 0..2047
        int r  = e >> 7, c = e & 127;
        float4 v = Xv[e4];
        As[r * LDS_STRIDE + c + 0] = v.x;
        As[r * LDS_STRIDE + c + 1] = v.y;
        As[r * LDS_STRIDE + c + 2] = v.z;
        As[r * LDS_STRIDE + c + 3] = v.w;
    }
    __syncthreads();

    const int lane = t & 31;
    const int wv   = t >> 5;
    const int half = lane >> 4;                    // 0: lanes 0-15, 1: lanes 16-31
    const int m    = lane & 15;
    const int col  = (wv << 4) + m;                // output column for this lane

    v8f c = {};
#pragma unroll 8
    for (int k0 = 0; k0 < D; k0 += 4) {
        const int kk = k0 + (half << 1);           // K=0/1 (lanes 0-15), K=2/3 (lanes 16-31)
        v2f a, b;
        a.x = As[m * LDS_STRIDE + kk];
        a.y = As[m * LDS_STRIDE + kk + 1];
        b.x = W[(size_t)kk * D + col];
        b.y = W[(size_t)(kk + 1) * D + col];
        c = __builtin_amdgcn_wmma_f32_16x16x4_f32(false, a, false, b, (short)0, c,
                                                  false, false);
    }

    const float bb = bias[col];
#pragma unroll
    for (int v = 0; v < 8; ++v) {                  // VGPR v holds row M = v + 8*half
        const int row = row0 + v + (half << 3);
        float r = c[v] + bb;
        Y[(size_t)row * D + col] = r > 0.0f ? r : 0.0f;
    }
}

// ---------------------------------------------------------------------------
// Fused SAGE combine: H2 = relu(H @ Wself + HN @ Wneigh + b)
// Same tiling as above, two A panels in LDS, two wmma per K-step into one acc.
// ---------------------------------------------------------------------------
__global__ __launch_bounds__(256) void sage_combine_wmma(
    const float* __restrict__ H, const float* __restrict__ HN,
    const float* __restrict__ Wself, const float* __restrict__ Wneigh,
    const float* __restrict__ bias, float* __restrict__ Y, int N)
{
    __shared__ float As[16 * LDS_STRIDE];
    __shared__ float An[16 * LDS_STRIDE];

    const int row0 = blockIdx.x << 4;
    const int t    = threadIdx.x;

    const float4* Hv = (const float4*)(H  + (size_t)row0 * D);
    const float4* Nv = (const float4*)(HN + (size_t)row0 * D);
#pragma unroll
    for (int i = 0; i < 2; ++i) {
        int e4 = t * 2 + i;
        int e  = e4 * 4;
        int r  = e >> 7, c = e & 127;
        float4 v = Hv[e4];
        As[r * LDS_STRIDE + c + 0] = v.x;
        As[r * LDS_STRIDE + c + 1] = v.y;
        As[r * LDS_STRIDE + c + 2] = v.z;
        As[r * LDS_STRIDE + c + 3] = v.w;
        float4 u = Nv[e4];
        An[r * LDS_STRIDE + c + 0] = u.x;
        An[r * LDS_STRIDE + c + 1] = u.y;
        An[r * LDS_STRIDE + c + 2] = u.z;
        An[r * LDS_STRIDE + c + 3] = u.w;
    }
    __syncthreads();

    const int lane = t & 31;
    const int wv   = t >> 5;
    const int half = lane >> 4;
    const int m    = lane & 15;
    const int col  = (wv << 4) + m;

    v8f c = {};
#pragma unroll 4
    for (int k0 = 0; k0 < D; k0 += 4) {
        const int kk = k0 + (half << 1);
        v2f a1, b1, a2, b2;
        a1.x = As[m * LDS_STRIDE + kk];
        a1.y = As[m * LDS_STRIDE + kk + 1];
        b1.x = Wself[(size_t)kk * D + col];
        b1.y = Wself[(size_t)(kk + 1) * D + col];
        c = __builtin_amdgcn_wmma_f32_16x16x4_f32(false, a1, false, b1, (short)0, c,
                                                  false, false);
        a2.x = An[m * LDS_STRIDE + kk];
        a2.y = An[m * LDS_STRIDE + kk + 1];
        b2.x = Wneigh[(size_t)kk * D + col];
        b2.y = Wneigh[(size_t)(kk + 1) * D + col];
        c = __builtin_amdgcn_wmma_f32_16x16x4_f32(false, a2, false, b2, (short)0, c,
                                                  false, false);
    }

    const float bb = bias[col];
#pragma unroll
    for (int v = 0; v < 8; ++v) {
        const int row = row0 + v + (half << 3);
        float r = c[v] + bb;
        Y[(size_t)row * D + col] = r > 0.0f ? r : 0.0f;
    }
}

// ---------------------------------------------------------------------------
// Edge scatter: one wave per edge. Lane reads float4 of h[src], atomically
// adds into nsum[dst]; lane 0 bumps degree. h/nsum are L2-resident (51 MB
// each vs 192 MB L2), so this runs at L2 atomic rate.
// ---------------------------------------------------------------------------
__global__ __launch_bounds__(256) void sage_edge_scatter(
    const float* __restrict__ h, const int* __restrict__ esrc,
    const int* __restrict__ edst, float* __restrict__ nsum,
    float* __restrict__ deg, int E)
{
    const int wave = (int)((blockIdx.x * blockDim.x + threadIdx.x) >> 5);
    const int lane = threadIdx.x & 31;
    if (wave >= E) return;

    const int s = esrc[wave];
    const int d = edst[wave];

    const float4 v = ((const float4*)(h + (size_t)s * D))[lane];  // 32 lanes x 16B = 512B
    float* np = nsum + (size_t)d * D + lane * 4;
    atomicAdd(np + 0, v.x);
    atomicAdd(np + 1, v.y);
    atomicAdd(np + 2, v.z);
    atomicAdd(np + 3, v.w);
    if (lane == 0) atomicAdd(deg + d, 1.0f);
}

// ---------------------------------------------------------------------------
// In-place mean: nsum[i][c] /= max(deg[i], 1)
// ---------------------------------------------------------------------------
__global__ __launch_bounds__(256) void sage_mean_div(
    float* __restrict__ nsum, const float* __restrict__ deg, long long total)
{
    long long i = (long long)blockIdx.x * blockDim.x + threadIdx.x;
    if (i < total) {
        float dg = deg[i >> 7];
        nsum[i] = nsum[i] / fmaxf(dg, 1.0f);
    }
}

// ---------------------------------------------------------------------------
extern "C" void kernel_launch(void* const* d_in, const int* in_sizes, int n_in,
                              void* d_out, int out_size, void* d_ws, size_t ws_size,
                              hipStream_t stream) {
    (void)n_in; (void)out_size; (void)ws_size;

    const float* x      = (const float*)d_in[0];
    const int*   esrc   = (const int*)d_in[1];
    const int*   edst   = (const int*)d_in[2];
    const float* W1     = (const float*)d_in[3];
    const float* b1     = (const float*)d_in[4];
    const float* Wself  = (const float*)d_in[5];
    const float* Wneigh = (const float*)d_in[6];
    const float* bsage  = (const float*)d_in[7];
    const float* W2     = (const float*)d_in[8];
    const float* b2     = (const float*)d_in[9];

    const int N = in_sizes[0] / D;   // 100000
    const int E = in_sizes[1];       // 1600000

    // Workspace layout (fp32): h[N*128] | nsum[N*128] | h2[N*128] | deg[N]
    float* h    = (float*)d_ws;
    float* nsum = h    + (size_t)N * D;
    float* h2   = nsum + (size_t)N * D;
    float* deg  = h2   + (size_t)N * D;

    const long long nd = (long long)N * D;

    // 1) zero accumulators
    sage_zero_f32<<<(unsigned)((nd + 255) / 256), 256, 0, stream>>>(nsum, nd);
    sage_zero_f32<<<(unsigned)((N + 255) / 256), 256, 0, stream>>>(deg, N);

    // 2) h = relu(x @ W1 + b1)
    sage_dense_relu_wmma<<<N / 16, 256, 0, stream>>>(x, W1, b1, h, N);

    // 3) edge gather + scatter-add (one wave per edge)
    sage_edge_scatter<<<(E + 7) / 8, 256, 0, stream>>>(h, esrc, edst, nsum, deg, E);

    // 4) nsum -> mean (in place)
    sage_mean_div<<<(unsigned)((nd + 255) / 256), 256, 0, stream>>>(nsum, deg, nd);

    // 5) h2 = relu(h @ Wself + nsum @ Wneigh + b_sage)  (fused dual GEMM)
    sage_combine_wmma<<<N / 16, 256, 0, stream>>>(h, nsum, Wself, Wneigh, bsage, h2, N);

    // 6) out = relu(h2 @ W2 + b2)
    sage_dense_relu_wmma<<<N / 16, 256, 0, stream>>>(h2, W2, b2, (float*)d_out, N);
}